// PhaseAttention_24043226923316
// MI455X (gfx1250) — compile-verified
//
#include <hip/hip_runtime.h>
#include <math.h>

typedef __attribute__((ext_vector_type(2))) float v2f;
typedef __attribute__((ext_vector_type(8))) float v8f;

#define PI_F      3.14159265358979323846f
#define TWO_PI_F  6.28318530717958647692f
#define INV_2PI_F 0.15915494309189533577f
#define DT_F      0.1f

// ---------------------------------------------------------------------------
// GEMM: out[M,512] = act(A[M,512] @ W[512,512]^T + bias[512])
//   mode 0 -> tanh(.)*pi   (phase head)
//   mode 1 -> identity     (output projection)
// One 16x64 output tile per wave: 4 accumulators share each A fragment.
// K fully unrolled; all load addresses are base + compile-time immediates so
// the backend can clause-group loads and stagger s_wait_loadcnt.
// ---------------------------------------------------------------------------
__global__ __launch_bounds__(256) void wmma_gemm_f32_kernel(
    const float* __restrict__ A, const float* __restrict__ W,
    const float* __restrict__ bias, float* __restrict__ out,
    int M, int mode)
{
    constexpr int N = 512;
    constexpr int K = 512;

    const int wave = threadIdx.x >> 5;
    const int lane = threadIdx.x & 31;
    const int hi   = lane >> 4;     // wave half (0/1)
    const int lrow = lane & 15;

    constexpr int ctiles = N >> 6;             // column groups of 64
    const int tile = blockIdx.x * 8 + wave;
    const int tm   = tile / ctiles;
    const int tc   = tile % ctiles;
    if (tm * 16 >= M) return;                  // wave-uniform exit

    // A fragment: lane l -> A[tm*16 + (l&15), k + 2*(l>>4) + {0,1}]
    // B fragment j: lane l -> W[tc*64 + j*16 + (l&15), k + 2*(l>>4) + {0,1}]
    // Single B base pointer; j*16 rows become immediate offsets (j*16*K floats).
    const float* arow = A + (size_t)(tm * 16 + lrow) * K + 2 * hi;
    const float* brow = W + (size_t)(tc * 64 + lrow) * K + 2 * hi;

    v8f acc0 = {}, acc1 = {}, acc2 = {}, acc3 = {};

#pragma unroll
    for (int k0 = 0; k0 < K; k0 += 16) {
        // phase 1: 20 fragment loads, all constant offsets from two pointers
        v2f a[4], b0[4], b1[4], b2[4], b3[4];
#pragma unroll
        for (int i = 0; i < 4; ++i) {
            const int k = k0 + 4 * i;
            a[i]  = *(const v2f*)(arow + k);
            b0[i] = *(const v2f*)(brow + k);
            b1[i] = *(const v2f*)(brow + k + 16 * K);
            b2[i] = *(const v2f*)(brow + k + 32 * K);
            b3[i] = *(const v2f*)(brow + k + 48 * K);
        }
        // phase 2: 16 WMMAs, 4 independent accumulator chains
#pragma unroll
        for (int i = 0; i < 4; ++i) {
            acc0 = __builtin_amdgcn_wmma_f32_16x16x4_f32(
                       false, a[i], false, b0[i], (short)0, acc0, false, false);
            acc1 = __builtin_amdgcn_wmma_f32_16x16x4_f32(
                       false, a[i], false, b1[i], (short)0, acc1, false, false);
            acc2 = __builtin_amdgcn_wmma_f32_16x16x4_f32(
                       false, a[i], false, b2[i], (short)0, acc2, false, false);
            acc3 = __builtin_amdgcn_wmma_f32_16x16x4_f32(
                       false, a[i], false, b3[i], (short)0, acc3, false, false);
        }
    }

    // C/D layout: VGPR v, lane<16 -> row v ; lane>=16 -> row v+8 ; col = lane&15
    const int row0 = tm * 16 + 8 * hi;
#pragma unroll
    for (int j = 0; j < 4; ++j) {
        const v8f acc = (j == 0) ? acc0 : (j == 1) ? acc1 : (j == 2) ? acc2 : acc3;
        const int col = tc * 64 + j * 16 + lrow;
        const float bi = bias[col];
#pragma unroll
        for (int v = 0; v < 8; ++v) {
            float r = acc[v] + bi;
            if (mode == 0) r = tanhf(r) * PI_F;
            out[(size_t)(row0 + v) * N + col] = r;
        }
    }
}

// ---------------------------------------------------------------------------
// Kuramoto synchronization: 10 fused iterations.
// Oscillators are independent across the column mean, so split each (b,h)
// over 4 blocks of 16 oscillators: 64 blocks x 512 threads.
// Thread (ol = tid&15, g = tid>>4) owns the 16 phases of oscillator
// o = oc*16+ol for rows s in [g*16, g*16+16), kept in registers.
// Masked sin/cos column means reduce through LDS (32 partials x 16 osc).
// ---------------------------------------------------------------------------
__global__ __launch_bounds__(512) void kuramoto_sync_kernel(
    float* __restrict__ P,                 // (1024, 512) phases, in/out
    const float* __restrict__ mask,        // (2, 512)
    const float* __restrict__ natural_freq,// (8, 64)
    const float* __restrict__ coupling)    // (8,1,1) -> K per head
{
    __shared__ float red_s[32][16];
    __shared__ float red_c[32][16];
    __shared__ float ms_sh[16];
    __shared__ float mc_sh[16];
    __shared__ float mask_sh[512];

    const int bh  = blockIdx.x >> 2;       // 0..15
    const int oc  = blockIdx.x & 3;        // oscillator chunk 0..3
    const int b   = bh >> 3;
    const int h   = bh & 7;
    const int tid = threadIdx.x;
    const int ol  = tid & 15;              // local oscillator
    const int g   = tid >> 4;              // row group 0..31
    const int o   = oc * 16 + ol;

    mask_sh[tid] = mask[b * 512 + tid];

    const float omega = natural_freq[h * 64 + o];
    const float Kc    = coupling[h];

    float* base = P + (size_t)(b * 512 + g * 16) * 512 + h * 64 + o;

    float ph[16];
#pragma unroll
    for (int r = 0; r < 16; ++r) ph[r] = base[(size_t)r * 512];

    __syncthreads();

    for (int it = 0; it < 10; ++it) {
        // pass 1: masked partial sums of sin/cos over this thread's 16 rows
        float ssum = 0.0f, csum = 0.0f;
#pragma unroll
        for (int r = 0; r < 16; ++r) {
            float s, c;
            __sincosf(ph[r], &s, &c);
            const float m = mask_sh[g * 16 + r];
            ssum = fmaf(m, s, ssum);
            csum = fmaf(m, c, csum);
        }
        red_s[g][ol] = ssum;
        red_c[g][ol] = csum;
        __syncthreads();

        if (g == 0) {
            float ts = 0.0f, tc = 0.0f;
#pragma unroll
            for (int gg = 0; gg < 32; ++gg) { ts += red_s[gg][ol]; tc += red_c[gg][ol]; }
            ms_sh[ol] = ts * (1.0f / 512.0f);   // jnp.mean divides by S
            mc_sh[ol] = tc * (1.0f / 512.0f);
        }
        __syncthreads();

        const float ms = ms_sh[ol];
        const float mc = mc_sh[ol];

        // pass 2: update + wrap to [-pi, pi) with positive mod (jnp.mod semantics)
#pragma unroll
        for (int r = 0; r < 16; ++r) {
            float s, c;
            __sincosf(ph[r], &s, &c);
            const float dtheta = omega + Kc * (c * ms - s * mc);
            float x = ph[r] + DT_F * dtheta + PI_F;
            x = x - TWO_PI_F * floorf(x * INV_2PI_F);
            ph[r] = x - PI_F;
        }
        // red_* rewrites next iteration are ordered by the mid barrier;
        // ms_sh reads above are ordered by the second barrier.
    }

#pragma unroll
    for (int r = 0; r < 16; ++r) base[(size_t)r * 512] = ph[r];
}

// ---------------------------------------------------------------------------
// inputs: x, mask, W_phase, b_phase, natural_freq, coupling_matrix, W_out, b_out
// ---------------------------------------------------------------------------
extern "C" void kernel_launch(void* const* d_in, const int* in_sizes, int n_in,
                              void* d_out, int out_size, void* d_ws, size_t ws_size,
                              hipStream_t stream) {
    const float* x            = (const float*)d_in[0];
    const float* mask         = (const float*)d_in[1];
    const float* W_phase      = (const float*)d_in[2];
    const float* b_phase      = (const float*)d_in[3];
    const float* natural_freq = (const float*)d_in[4];
    const float* coupling     = (const float*)d_in[5];
    const float* W_out        = (const float*)d_in[6];
    const float* b_out        = (const float*)d_in[7];
    float*       out          = (float*)d_out;
    float*       P            = (float*)d_ws;     // 1024 x 512 fp32 = 2 MB scratch

    const int D = 512;                  // D_MODEL == H*O == N == K
    const int M = in_sizes[0] / D;      // B*S = 1024
    const int waves = (M / 16) * (D / 64);   // 512 waves, 8 per block

    dim3 block(256);
    dim3 grid(waves / 8);               // 64 blocks

    // 1) phases = tanh(x @ W_phase^T + b_phase) * pi   -> P
    wmma_gemm_f32_kernel<<<grid, block, 0, stream>>>(x, W_phase, b_phase, P, M, 0);
    // 2) 10 Kuramoto iterations, fused, phase state register-resident
    kuramoto_sync_kernel<<<dim3(64), dim3(512), 0, stream>>>(P, mask,
                                                             natural_freq, coupling);
    // 3) out = P @ W_out^T + b_out
    wmma_gemm_f32_kernel<<<grid, block, 0, stream>>>(P, W_out, b_out, out, M, 1);
}